// NolinerSEM_5420248727820
// MI455X (gfx1250) — compile-verified
//
#include <hip/hip_runtime.h>

#define B_TOTAL 32768
#define E_DIM 90
#define H_DIM 32
#define NEG_SLOPE 0.33f
#define K_PAD 92            // E rounded up to multiple of 4 (WMMA f32 K-step = 4)
#define K_STEPS 23          // 92 / 4
#define WAVES_PER_BLOCK 8
#define TILES_PER_WAVE 4
#define ROWS_PER_BLOCK (WAVES_PER_BLOCK * TILES_PER_WAVE * 16)   // 512

typedef float v2f __attribute__((ext_vector_type(2)));
typedef float v8f __attribute__((ext_vector_type(8)));

__device__ __forceinline__ float leaky(float v) {
    // branchless: v>=0 -> v ; v<0 -> NEG_SLOPE*v
    return fmaxf(v, 0.0f) + NEG_SLOPE * fminf(v, 0.0f);
}

__global__ __launch_bounds__(256) void fused_mlp_wmma(
    const float* __restrict__ x,    // [B, 1, E, E]
    const float* __restrict__ W1,   // [E, H, E]
    const float* __restrict__ b1,   // [E, H]
    const float* __restrict__ W2,   // [E, 1, H]
    const float* __restrict__ b2,   // [E, 1]
    float* __restrict__ out)        // [B, E]
{
    __shared__ __align__(16) float w1s[H_DIM * K_PAD];  // [h][f], f padded 90..91 = 0

    const int e   = blockIdx.x;
    const int tid = threadIdx.x;

    // Stage W1[e,:,:] into LDS, zero-padding K to 92.
    for (int idx = tid; idx < H_DIM * K_PAD; idx += 256) {
        const int h = idx / K_PAD;
        const int f = idx - h * K_PAD;
        w1s[idx] = (f < E_DIM) ? W1[(e * H_DIM + h) * E_DIM + f] : 0.0f;
    }
    __syncthreads();

    const int lane = tid & 31;
    const int wave = tid >> 5;
    const int half = lane >> 4;     // A/B fragment half: K{0,1} vs K{2,3}
    const int n    = lane & 15;     // column index within a 16-wide N tile
    const int koff = half * 2;

    // Per-lane h-dependent constants (h = n for tile0, n+16 for tile1).
    const float b1a   = b1[e * H_DIM + n];
    const float b1b   = b1[e * H_DIM + n + 16];
    const float w2a   = W2[e * H_DIM + n];
    const float w2b   = W2[e * H_DIM + n + 16];
    const float bias2 = b2[e];

    const int firstTile = blockIdx.y * (WAVES_PER_BLOCK * TILES_PER_WAVE)
                        + wave * TILES_PER_WAVE;

    for (int t = 0; t < TILES_PER_WAVE; ++t) {
        const int rowBase = (firstTile + t) * 16;
        const int row     = rowBase + n;                  // both halves: M = n (+8 via C layout)
        const float* __restrict__ xrow =
            x + (size_t)row * (E_DIM * E_DIM) + (size_t)e * E_DIM;

        if (t + 1 < TILES_PER_WAVE)
            __builtin_prefetch(xrow + 16 * E_DIM * E_DIM, 0, 1);  // global_prefetch_b8

        // Preload all A fragments: lane holds x[row, 4j+koff .. +1].
        v2f a[K_STEPS];
#pragma unroll
        for (int j = 0; j < K_STEPS; ++j) {
            int f = 4 * j + koff;
            if (f > E_DIM - 2) f = E_DIM - 2;   // clamp tail; padded B rows are zero
            a[j][0] = xrow[f];
            a[j][1] = xrow[f + 1];
        }

        v8f c0 = {};   // N = 0..15
        v8f c1 = {};   // N = 16..31
#pragma unroll
        for (int j = 0; j < K_STEPS; ++j) {
            const int f = 4 * j + koff;
            const v2f bf0 = *(const v2f*)&w1s[n * K_PAD + f];
            const v2f bf1 = *(const v2f*)&w1s[(n + 16) * K_PAD + f];
            c0 = __builtin_amdgcn_wmma_f32_16x16x4_f32(false, a[j], false, bf0,
                                                       (short)0, c0, false, false);
            c1 = __builtin_amdgcn_wmma_f32_16x16x4_f32(false, a[j], false, bf1,
                                                       (short)0, c1, false, false);
        }

        // Epilogue: bias + leaky, then W2-weighted reduce over the 32 h values.
        float partial[8];
#pragma unroll
        for (int r = 0; r < 8; ++r) {
            const float h0 = leaky(c0[r] + b1a);
            const float h1 = leaky(c1[r] + b1b);
            partial[r] = h0 * w2a + h1 * w2b;
        }
        // 16-lane butterfly; lanes 0-15 carry rows M=r, lanes 16-31 rows M=r+8.
#pragma unroll
        for (int r = 0; r < 8; ++r) {
#pragma unroll
            for (int m = 1; m < 16; m <<= 1)
                partial[r] += __shfl_xor(partial[r], m, 32);
        }
        if (n == 0) {   // lane 0 -> rows rowBase+0..7, lane 16 -> rowBase+8..15
#pragma unroll
            for (int r = 0; r < 8; ++r) {
                out[(size_t)(rowBase + 8 * half + r) * E_DIM + e] =
                    leaky(partial[r] + bias2);
            }
        }
    }
}

extern "C" void kernel_launch(void* const* d_in, const int* in_sizes, int n_in,
                              void* d_out, int out_size, void* d_ws, size_t ws_size,
                              hipStream_t stream) {
    const float* x  = (const float*)d_in[0];
    const float* W1 = (const float*)d_in[1];
    const float* b1 = (const float*)d_in[2];
    const float* W2 = (const float*)d_in[3];
    const float* b2 = (const float*)d_in[4];
    float* out = (float*)d_out;

    dim3 grid(E_DIM, B_TOTAL / ROWS_PER_BLOCK);   // (90, 64)
    fused_mlp_wmma<<<grid, dim3(256), 0, stream>>>(x, W1, b1, W2, b2, out);
}